// CustomEncoderDecoder_Attention_34729105555912
// MI455X (gfx1250) — compile-verified
//
#include <hip/hip_runtime.h>
#include <hip/hip_bf16.h>
#include <math.h>

// Problem constants (fixed by the reference).
#define S_LEN 512
#define I_DIM 256
#define H_DIM 1024
#define O_DIM 256
#define T_STEPS 64
#define G4H 4096   // 4*H

typedef float v2f __attribute__((ext_vector_type(2)));
typedef float v8f __attribute__((ext_vector_type(8)));

__device__ __forceinline__ float sigf(float x) { return 1.0f / (1.0f + __expf(-x)); }

// ---------------------------------------------------------------------------
// Setup kernels (one-shot, stream-ordered)
// ---------------------------------------------------------------------------

// K-packed-by-4 layout for matvec weights:
//   dst[ ((k>>2)*R + r)*4 + (k&3) ] = src[ r*srcStride + srcOff + k ]
// so a thread owning output row r reads float4 dst4[kb*R + r] = {k=4kb..4kb+3}.
__global__ void k_pack4(const float* __restrict__ src, float* __restrict__ dst,
                        int R, int Kn, int srcStride, int srcOff) {
  size_t total = (size_t)R * Kn;
  for (size_t idx = (size_t)blockIdx.x * blockDim.x + threadIdx.x; idx < total;
       idx += (size_t)gridDim.x * blockDim.x) {
    size_t r = idx / Kn, k = idx % Kn;
    dst[((k >> 2) * (size_t)R + r) * 4 + (k & 3)] =
        src[r * (size_t)srcStride + srcOff + k];
  }
}

__global__ void k_addvec(const float* __restrict__ a, const float* __restrict__ b,
                         float* __restrict__ dst, int n) {
  int i = blockIdx.x * blockDim.x + threadIdx.x;
  if (i < n) dst[i] = a[i] + b[i];
}

__global__ void k_zero(float* __restrict__ dst, int n) {
  int i = blockIdx.x * blockDim.x + threadIdx.x;
  if (i < n) dst[i] = 0.0f;
}

// gconst[row] = dec_b_ih[row] + dec_b_hh[row] + sum_k dec_W_ih[row,k] * b_o2h[k], k<H
__global__ void k_dec_const(const float* __restrict__ dec_W_ih,
                            const float* __restrict__ b_ih, const float* __restrict__ b_hh,
                            const float* __restrict__ b_o2h, float* __restrict__ gconst) {
  __shared__ float emb[H_DIM];
  for (int i = threadIdx.x; i < H_DIM; i += blockDim.x) emb[i] = b_o2h[i];
  __syncthreads();
  int row = blockIdx.x * blockDim.x + threadIdx.x;
  if (row >= G4H) return;
  float acc = b_ih[row] + b_hh[row];
  const float* wr = dec_W_ih + (size_t)row * (2 * H_DIM);
  for (int k = 0; k < H_DIM; ++k) acc += wr[k] * emb[k];
  gconst[row] = acc;
}

// ---------------------------------------------------------------------------
// WMMA fp32 GEMM:  D[M,N] = A[M,K] @ Bt[N,K]^T + bias[N]
// One wave32 per 32x32 output tile (2x2 accumulators of 16x16) so A/B
// fragments are each reused twice.  V_WMMA_F32_16X16X4_F32 over K.
// Frag layout: lanes 0-15 -> row=lane, K=k0,k0+1 ; lanes 16-31 -> row=lane-16,
// K=k0+2,k0+3.  D: vgpr r -> M = r + 8*(lane>=16), N = lane&15.
// ---------------------------------------------------------------------------
__global__ void k_gemm_wmma_f32(const float* __restrict__ A, const float* __restrict__ Bt,
                                const float* __restrict__ bias, float* __restrict__ D,
                                int M, int N, int K) {
  int lane = threadIdx.x & 31;
  int wid = (int)((blockIdx.x * (size_t)blockDim.x + threadIdx.x) >> 5);
  int ntn = N >> 5;                 // 32-wide tiles along N
  int tm = wid / ntn;
  int tn = wid - tm * ntn;
  if (tm >= (M >> 5)) return;       // wave-uniform guard; EXEC stays all-ones

  int frow = lane & 15;             // fragment row (A: M, B: N)
  int khi  = (lane >> 4) << 1;      // 0 or 2
  const float* a0 = A  + (size_t)(tm * 32 + frow) * K + khi;
  const float* a1 = a0 + (size_t)16 * K;
  const float* b0 = Bt + (size_t)(tn * 32 + frow) * K + khi;
  const float* b1 = b0 + (size_t)16 * K;

  v8f c00 = {}, c01 = {}, c10 = {}, c11 = {};
  for (int k = 0; k < K; k += 4) {
    v2f A0 = *(const v2f*)(a0 + k);
    v2f A1 = *(const v2f*)(a1 + k);
    v2f B0 = *(const v2f*)(b0 + k);
    v2f B1 = *(const v2f*)(b1 + k);
    c00 = __builtin_amdgcn_wmma_f32_16x16x4_f32(false, A0, false, B0, (short)0, c00, false, false);
    c01 = __builtin_amdgcn_wmma_f32_16x16x4_f32(false, A0, false, B1, (short)0, c01, false, false);
    c10 = __builtin_amdgcn_wmma_f32_16x16x4_f32(false, A1, false, B0, (short)0, c10, false, false);
    c11 = __builtin_amdgcn_wmma_f32_16x16x4_f32(false, A1, false, B1, (short)0, c11, false, false);
  }

  int n = lane & 15;
  int mb = (lane >> 4) << 3;        // 0 or 8
  float bn0 = bias[tn * 32 + n];
  float bn1 = bias[tn * 32 + 16 + n];
  float* d0 = D + (size_t)(tm * 32 + mb) * N + tn * 32 + n;
  float* d1 = D + (size_t)(tm * 32 + 16 + mb) * N + tn * 32 + n;
#pragma unroll
  for (int r = 0; r < 8; ++r) {
    d0[(size_t)r * N]      = c00[r] + bn0;
    d0[(size_t)r * N + 16] = c01[r] + bn1;
    d1[(size_t)r * N]      = c10[r] + bn0;
    d1[(size_t)r * N + 16] = c11[r] + bn1;
  }
}

// ---------------------------------------------------------------------------
// Encoder recurrence step: one launch per t, 32 blocks x 256 threads.
// Block b owns hidden units [32b, 32b+32): 128 gate rows (4 gates x 32 units),
// each a 1024-dot against h (LDS float4) with K-packed W4 -> one b128 global
// load per 4 MACs, fully coalesced per wave32.  Then LSTM pointwise update.
// hin/hout ping-pong through O_e rows; c is in-place (block-private units).
// ---------------------------------------------------------------------------
__global__ void k_enc_step(const float4* __restrict__ W4,  // [256][4096] float4
                           const float* __restrict__ xg,   // [4H] row for this t
                           const float* __restrict__ hin,
                           float* __restrict__ hout,       // = O_e row t
                           float* __restrict__ c) {
  __shared__ float4 sh4[H_DIM / 4];
  __shared__ float psum[128];
  __shared__ float gates[128];
  int tid = threadIdx.x;
  sh4[tid] = ((const float4*)hin)[tid];   // 256 threads == 256 float4
  __syncthreads();

  int half = tid >> 7;      // which half of K
  int lr   = tid & 127;     // local gate-row
  int gate = lr >> 5;
  int ul   = lr & 31;
  int row  = gate * H_DIM + blockIdx.x * 32 + ul;

  float acc = 0.0f;
  int kb0 = half * 128;
  for (int kb = 0; kb < 128; ++kb) {
    float4 w = W4[(size_t)(kb0 + kb) * G4H + row];
    float4 hv = sh4[kb0 + kb];
    acc += hv.x * w.x + hv.y * w.y + hv.z * w.z + hv.w * w.w;
  }

  if (half) psum[lr] = acc;
  __syncthreads();
  if (!half) gates[lr] = acc + psum[lr] + xg[row];
  __syncthreads();

  if (tid < 32) {
    int j = blockIdx.x * 32 + tid;
    float gi = sigf(gates[tid]);
    float gf = sigf(gates[32 + tid]);
    float gg = tanhf(gates[64 + tid]);
    float go = sigf(gates[96 + tid]);
    float cn = gf * c[j] + gi * gg;
    float hn = go * tanhf(cn);
    c[j] = cn;
    hout[j] = hn;
  }
}

// Decoder LSTM step: gates = gconst + Whh@h + Wix@ctx, then cell update.
__global__ void k_dec_step(const float4* __restrict__ Whh4,  // [256][4096] float4
                           const float4* __restrict__ Wix4,  // [256][4096] float4
                           const float* __restrict__ gconst,
                           const float* __restrict__ ctx,
                           const float* __restrict__ hin,
                           float* __restrict__ hout, float* __restrict__ c) {
  __shared__ float4 sh4[H_DIM / 4];
  __shared__ float4 sc4[H_DIM / 4];
  __shared__ float psum[128];
  __shared__ float gates[128];
  int tid = threadIdx.x;
  sh4[tid] = ((const float4*)hin)[tid];
  sc4[tid] = ((const float4*)ctx)[tid];
  __syncthreads();

  int half = tid >> 7;
  int lr   = tid & 127;
  int gate = lr >> 5;
  int ul   = lr & 31;
  int row  = gate * H_DIM + blockIdx.x * 32 + ul;

  float acc = 0.0f;
  int kb0 = half * 128;
  for (int kb = 0; kb < 128; ++kb) {
    size_t ko = (size_t)(kb0 + kb) * G4H + row;
    float4 wh = Whh4[ko];
    float4 wc = Wix4[ko];
    float4 hv = sh4[kb0 + kb];
    float4 cv = sc4[kb0 + kb];
    acc += hv.x * wh.x + hv.y * wh.y + hv.z * wh.z + hv.w * wh.w;
    acc += cv.x * wc.x + cv.y * wc.y + cv.z * wc.z + cv.w * wc.w;
  }
  if (half) psum[lr] = acc;
  __syncthreads();
  if (!half) gates[lr] = acc + psum[lr] + gconst[row];
  __syncthreads();

  if (tid < 32) {
    int j = blockIdx.x * 32 + tid;
    float gi = sigf(gates[tid]);
    float gf = sigf(gates[32 + tid]);
    float gg = tanhf(gates[64 + tid]);
    float go = sigf(gates[96 + tid]);
    float cn = gf * c[j] + gi * gg;
    float hn = go * tanhf(cn);
    c[j] = cn;
    hout[j] = hn;
  }
}

// dw = h @ W_dw^T + b_dw   (K-packed Wdw4, 4 blocks x 256, one thread per output)
__global__ void k_dec_dw(const float4* __restrict__ Wdw4, const float* __restrict__ b_dw,
                         const float* __restrict__ hin, float* __restrict__ dw) {
  __shared__ float4 sh4[H_DIM / 4];
  int tid = threadIdx.x;
  sh4[tid] = ((const float4*)hin)[tid];
  __syncthreads();
  int j = blockIdx.x * 256 + tid;
  float acc = b_dw[j];
  for (int kb = 0; kb < 256; ++kb) {
    float4 w = Wdw4[(size_t)kb * H_DIM + j];
    float4 hv = sh4[kb];
    acc += hv.x * w.x + hv.y * w.y + hv.z * w.z + hv.w * w.w;
  }
  dw[j] = acc;
}

// vi[s] = W_att . tanh(yu[s,:] + dw) + b_att ; attw = softmax(vi).  1 block x 512.
__global__ void k_dec_att(const float* __restrict__ yu, const float* __restrict__ dw,
                          const float* __restrict__ Watt, const float* __restrict__ b_att,
                          float* __restrict__ attw) {
  __shared__ float4 sdw4[H_DIM / 4];
  __shared__ float4 swt4[H_DIM / 4];
  __shared__ float red[S_LEN];
  int tid = threadIdx.x;
  if (tid < 256) {
    sdw4[tid] = ((const float4*)dw)[tid];
    swt4[tid] = ((const float4*)Watt)[tid];
  }
  __syncthreads();

  const float4* yr4 = (const float4*)(yu + (size_t)tid * H_DIM);
  float v = b_att[0];
  for (int kb = 0; kb < 256; ++kb) {
    float4 y = yr4[kb];
    float4 d = sdw4[kb];
    float4 w = swt4[kb];
    v += tanhf(y.x + d.x) * w.x + tanhf(y.y + d.y) * w.y +
         tanhf(y.z + d.z) * w.z + tanhf(y.w + d.w) * w.w;
  }

  red[tid] = v; __syncthreads();
  for (int s = 256; s > 0; s >>= 1) {
    if (tid < s) red[tid] = fmaxf(red[tid], red[tid + s]);
    __syncthreads();
  }
  float m = red[0]; __syncthreads();
  float e = __expf(v - m);
  red[tid] = e; __syncthreads();
  for (int s = 256; s > 0; s >>= 1) {
    if (tid < s) red[tid] += red[tid + s];
    __syncthreads();
  }
  attw[tid] = e / red[0];
}

// ctx[j0..j0+3] = sum_s attw[s] * O_e[s, j0..j0+3]   (1 block x 256, float4 cols)
__global__ void k_dec_ctx(const float* __restrict__ attw, const float* __restrict__ Oe,
                          float* __restrict__ ctx) {
  __shared__ float sw[S_LEN];
  int tid = threadIdx.x;
  sw[tid] = attw[tid];
  sw[tid + 256] = attw[tid + 256];
  __syncthreads();
  const float4* Oe4 = (const float4*)Oe;    // row s: Oe4[s*256 + tid]
  float4 acc = {0.f, 0.f, 0.f, 0.f};
  for (int s = 0; s < S_LEN; ++s) {
    float w = sw[s];
    float4 v = Oe4[(size_t)s * (H_DIM / 4) + tid];
    acc.x += w * v.x; acc.y += w * v.y; acc.z += w * v.z; acc.w += w * v.w;
  }
  ((float4*)ctx)[tid] = acc;
}

// out = log_softmax(h @ W_h2o^T + b_h2o)   (1 block x 256, K-packed Wh2o4)
__global__ void k_dec_out(const float4* __restrict__ Wh2o4, const float* __restrict__ b,
                          const float* __restrict__ h, float* __restrict__ out) {
  __shared__ float4 sh4[H_DIM / 4];
  __shared__ float red[O_DIM];
  int tid = threadIdx.x;
  sh4[tid] = ((const float4*)h)[tid];
  __syncthreads();
  float acc = b[tid];
  for (int kb = 0; kb < 256; ++kb) {
    float4 w = Wh2o4[(size_t)kb * O_DIM + tid];
    float4 hv = sh4[kb];
    acc += hv.x * w.x + hv.y * w.y + hv.z * w.z + hv.w * w.w;
  }

  red[tid] = acc; __syncthreads();
  for (int s = 128; s > 0; s >>= 1) {
    if (tid < s) red[tid] = fmaxf(red[tid], red[tid + s]);
    __syncthreads();
  }
  float m = red[0]; __syncthreads();
  float e = __expf(acc - m);
  red[tid] = e; __syncthreads();
  for (int s = 128; s > 0; s >>= 1) {
    if (tid < s) red[tid] += red[tid + s];
    __syncthreads();
  }
  out[tid] = (acc - m) - __logf(red[0]);
}

// ---------------------------------------------------------------------------
// Host-side orchestration
// ---------------------------------------------------------------------------
extern "C" void kernel_launch(void* const* d_in, const int* in_sizes, int n_in,
                              void* d_out, int out_size, void* d_ws, size_t ws_size,
                              hipStream_t stream) {
  const float* pt       = (const float*)d_in[0];
  const float* enc_W_ih = (const float*)d_in[1];
  const float* enc_W_hh = (const float*)d_in[2];
  const float* enc_b_ih = (const float*)d_in[3];
  const float* enc_b_hh = (const float*)d_in[4];
  const float* dec_W_ih = (const float*)d_in[5];
  const float* dec_W_hh = (const float*)d_in[6];
  const float* dec_b_ih = (const float*)d_in[7];
  const float* dec_b_hh = (const float*)d_in[8];
  const float* W_h2o    = (const float*)d_in[9];
  const float* b_h2o    = (const float*)d_in[10];
  const float* W_dw     = (const float*)d_in[11];
  const float* b_dw     = (const float*)d_in[12];
  const float* W_yu     = (const float*)d_in[13];
  const float* b_yu     = (const float*)d_in[14];
  const float* W_att    = (const float*)d_in[15];
  const float* b_att    = (const float*)d_in[16];
  // d_in[17] = W_o2h (unused in forward pass)
  const float* b_o2h    = (const float*)d_in[18];
  // d_in[19] = chars_otpt_max (fixed T=64)
  float* out = (float*)d_out;

  float* ws = (float*)d_ws;
  size_t o = 0;
  float* encWhh4 = ws + o; o += (size_t)H_DIM * G4H;      // 16 MB (K-packed x4)
  float* decWhh4 = ws + o; o += (size_t)H_DIM * G4H;      // 16 MB
  float* decWix4 = ws + o; o += (size_t)H_DIM * G4H;      // 16 MB (ctx half of W_ih)
  float* Wdw4    = ws + o; o += (size_t)H_DIM * H_DIM;    // 4 MB
  float* Wh2o4   = ws + o; o += (size_t)H_DIM * O_DIM;    // 1 MB
  float* xg      = ws + o; o += (size_t)S_LEN * G4H;      // 8 MB
  float* Oe      = ws + o; o += (size_t)S_LEN * H_DIM;    // 2 MB
  float* yu      = ws + o; o += (size_t)S_LEN * H_DIM;    // 2 MB
  float* encb    = ws + o; o += G4H;
  float* gconst  = ws + o; o += G4H;
  float* h0      = ws + o; o += H_DIM;
  float* hA      = ws + o; o += H_DIM;
  float* hB      = ws + o; o += H_DIM;
  float* cbuf    = ws + o; o += H_DIM;
  float* dw      = ws + o; o += H_DIM;
  float* ctx     = ws + o; o += H_DIM;
  float* attw    = ws + o; o += S_LEN;
  (void)ws_size; (void)in_sizes; (void)n_in; (void)out_size;

  // ---- one-shot setup: K-packed weight layouts + fused constants ----
  k_pack4<<<2048, 256, 0, stream>>>(enc_W_hh, encWhh4, G4H, H_DIM, H_DIM, 0);
  k_pack4<<<2048, 256, 0, stream>>>(dec_W_hh, decWhh4, G4H, H_DIM, H_DIM, 0);
  k_pack4<<<2048, 256, 0, stream>>>(dec_W_ih, decWix4, G4H, H_DIM, 2 * H_DIM, H_DIM);
  k_pack4<<<512, 256, 0, stream>>>(W_dw, Wdw4, H_DIM, H_DIM, H_DIM, 0);
  k_pack4<<<128, 256, 0, stream>>>(W_h2o, Wh2o4, O_DIM, H_DIM, H_DIM, 0);
  k_addvec<<<16, 256, 0, stream>>>(enc_b_ih, enc_b_hh, encb, G4H);
  k_dec_const<<<16, 256, 0, stream>>>(dec_W_ih, dec_b_ih, dec_b_hh, b_o2h, gconst);
  k_zero<<<4, 256, 0, stream>>>(h0, H_DIM);
  k_zero<<<4, 256, 0, stream>>>(cbuf, H_DIM);

  // ---- input-gate GEMM (WMMA f32): xg[512,4096] = pt @ enc_W_ih^T + encb ----
  // 32x32 tiles: (512/32)*(4096/32) = 2048 waves ; 8 waves/block -> 256 blocks
  k_gemm_wmma_f32<<<256, 256, 0, stream>>>(pt, enc_W_ih, encb, xg,
                                           S_LEN, G4H, I_DIM);

  // ---- encoder recurrence: 512 stream-ordered steps, h ping-pongs via O_e ----
  for (int t = 0; t < S_LEN; ++t) {
    const float* hin = (t == 0) ? h0 : (Oe + (size_t)(t - 1) * H_DIM);
    k_enc_step<<<32, 256, 0, stream>>>((const float4*)encWhh4,
                                       xg + (size_t)t * G4H, hin,
                                       Oe + (size_t)t * H_DIM, cbuf);
  }

  // ---- yu GEMM (WMMA f32): yu[512,1024] = O_e @ W_yu^T + b_yu ----
  // 32x32 tiles: 16*32 = 512 waves -> 64 blocks
  k_gemm_wmma_f32<<<64, 256, 0, stream>>>(Oe, W_yu, b_yu, yu,
                                          S_LEN, H_DIM, H_DIM);

  // ---- decoder: 64 steps of {dw, attention, ctx, LSTM, log_softmax} ----
  const float* hprev = Oe + (size_t)(S_LEN - 1) * H_DIM;  // h_enc
  for (int t = 0; t < T_STEPS; ++t) {
    float* hnext = (t & 1) ? hB : hA;
    k_dec_dw<<<4, 256, 0, stream>>>((const float4*)Wdw4, b_dw, hprev, dw);
    k_dec_att<<<1, 512, 0, stream>>>(yu, dw, W_att, b_att, attw);
    k_dec_ctx<<<1, 256, 0, stream>>>(attw, Oe, ctx);
    k_dec_step<<<32, 256, 0, stream>>>((const float4*)decWhh4, (const float4*)decWix4,
                                       gconst, ctx, hprev, hnext, cbuf);
    k_dec_out<<<1, 256, 0, stream>>>((const float4*)Wh2o4, b_h2o, hnext,
                                     out + (size_t)t * O_DIM);
    hprev = hnext;
  }
}